// SparseSpatialFluxComponent_14336600834356
// MI455X (gfx1250) — compile-verified
//
#include <hip/hip_runtime.h>

typedef float v2f __attribute__((ext_vector_type(2)));
typedef float v8f __attribute__((ext_vector_type(8)));

#define N_SRC 128
#define HH 1024
#define WW 1024
#define TILES_PER_ROW (WW / 16)        // 64
#define N_TILES ((HH / 16) * (WW / 16)) // 4096
#define WAVES_PER_BLOCK 8

// Out = A x B, K = 128:
//   A[h,n] = clamp(1 - |h - x_pos[n]|, 0, 1) * exp(log_flux[n])
//   B[n,w] = clamp(1 - |w - y_pos[n]|, 0, 1)
// One wave32 computes one 16x16 tile via 32 x v_wmma_f32_16x16x4_f32.
__global__ __launch_bounds__(256) void flux_splat_wmma(
    const float* __restrict__ log_flux,
    const float* __restrict__ x_pos,
    const float* __restrict__ y_pos,
    float* __restrict__ out)
{
    __shared__ float sX[N_SRC];   // x_pos (pairs with row index)
    __shared__ float sY[N_SRC];   // y_pos (pairs with col index)
    __shared__ float sE[N_SRC];   // exp(log_flux)

    const int t = threadIdx.x;
    if (t < N_SRC) {
        sX[t] = x_pos[t];
        sE[t] = expf(log_flux[t]);
    } else {
        const int i = t - N_SRC;
        sY[i] = y_pos[i];
    }
    __syncthreads();

    const int wave = t >> 5;
    const int lane = t & 31;
    const int m    = lane & 15;   // A: row-in-tile, B: col-in-tile
    const int half = lane >> 4;   // selects K sub-pair within each K=4 step

    const int tile    = blockIdx.x * WAVES_PER_BLOCK + wave;
    const int tileRow = tile >> 6;           // / TILES_PER_ROW
    const int tileCol = tile & (TILES_PER_ROW - 1);

    const float rowf = (float)(tileRow * 16 + m);
    const float colf = (float)(tileCol * 16 + m);

    v8f c = {0.f, 0.f, 0.f, 0.f, 0.f, 0.f, 0.f, 0.f};

#pragma unroll 4
    for (int k0 = 0; k0 < N_SRC; k0 += 4) {
        const int ka = k0 + 2 * half;  // this half-wave's K pair

        // A fragment: 16x4 f32, lanes 0-15 -> K={k0,k0+1}, lanes 16-31 -> K={k0+2,k0+3}
        v2f a, b;
        {
            const float x0 = sX[ka], x1 = sX[ka + 1];
            const float e0 = sE[ka], e1 = sE[ka + 1];
            const float wx0 = fmaxf(1.0f - fabsf(rowf - x0), 0.0f); // 1-|d| <= 1 always
            const float wx1 = fmaxf(1.0f - fabsf(rowf - x1), 0.0f);
            a.x = wx0 * e0;
            a.y = wx1 * e1;
        }
        // B fragment: 4x16 f32, mirrored layout (lane = N, half selects K pair)
        {
            const float y0v = sY[ka], y1v = sY[ka + 1];
            b.x = fmaxf(1.0f - fabsf(colf - y0v), 0.0f);
            b.y = fmaxf(1.0f - fabsf(colf - y1v), 0.0f);
        }

        // 8 args: (neg_a, A, neg_b, B, c_mod, C, reuse_a, reuse_b)
        c = __builtin_amdgcn_wmma_f32_16x16x4_f32(
                false, a, false, b, (short)0, c, false, false);
    }

    // C/D layout: VGPR r -> row = r (lanes 0-15) / row = 8+r (lanes 16-31), col = m
    const int colIdx  = tileCol * 16 + m;
    const int rowBase = tileRow * 16 + 8 * half;
#pragma unroll
    for (int r = 0; r < 8; ++r) {
        out[(rowBase + r) * WW + colIdx] = c[r];
    }
}

extern "C" void kernel_launch(void* const* d_in, const int* in_sizes, int n_in,
                              void* d_out, int out_size, void* d_ws, size_t ws_size,
                              hipStream_t stream) {
    const float* log_flux = (const float*)d_in[0];
    const float* x_pos    = (const float*)d_in[1];
    const float* y_pos    = (const float*)d_in[2];
    float* out = (float*)d_out;

    dim3 block(256);
    dim3 grid(N_TILES / WAVES_PER_BLOCK); // 512 blocks, 8 waves (tiles) each
    flux_splat_wmma<<<grid, block, 0, stream>>>(log_flux, x_pos, y_pos, out);
}